// KalmanFilter_67894843015848
// MI455X (gfx1250) — compile-verified
//
#include <hip/hip_runtime.h>
#include <hip/hip_bf16.h>
#include <math.h>

#define Gn 2048
#define Tn 256
#define Mn 8
#define Sn 16

typedef __attribute__((ext_vector_type(2))) float v2f;
typedef __attribute__((ext_vector_type(8))) float v8f;

// L0 element: lower-tri Cholesky factor from log-diag + strict-lower (row-major tril order)
__device__ __forceinline__ float l0elem(int i, int k, const float* ld, const float* od) {
    if (k > i) return 0.0f;
    if (k == i) return expf(ld[i]);
    return od[i * (i - 1) / 2 + k];
}

// ---------------------------------------------------------------------------
// Kernel 1: shared covariance recursion (P_t identical for all groups).
// Emits:
//   covTab[T][8][8]      : ycov_t = H P_t H^T + R   (== Smat_t)
//   Atab [T-1][4][32][2] : A_t = F (I - K^T H), pre-swizzled to WMMA-A layout
//   Btab [T-1][2][32][2] : B_t = F K^T,          pre-swizzled to WMMA-A layout
//   Hswz [4][32][2]      : H padded to 16x16,    pre-swizzled to WMMA-A layout
// WMMA-A layout for 16xK fp32 chunk q: lane l = 16h+n holds {M[l%16][4q+2h], M[l%16][4q+2h+1]}
// ---------------------------------------------------------------------------
__global__ void kf_precompute(const float* __restrict__ ilog, const float* __restrict__ ioff,
                              const float* __restrict__ F, const float* __restrict__ H,
                              const float* __restrict__ LQ, const float* __restrict__ LR,
                              float* __restrict__ covTab, float* __restrict__ Atab,
                              float* __restrict__ Btab, float* __restrict__ Hswz) {
    __shared__ float P[16][16], Pn[16][16], HP[8][16], Sm[8][8], Lc[8][8];
    __shared__ float Kt[8][16], FK[16][8], Qs[16][16], Rs[8][8];
    __shared__ float Fs[16][16], Hs[8][16], Pu[16][16], FPm[16][16], As[16][16];
    const int tid = threadIdx.x;
    const int i4 = tid >> 4, j4 = tid & 15;

    // constants
    Fs[i4][j4] = F[tid];
    if (tid < 128) Hs[tid >> 4][tid & 15] = H[tid];
    {   // Q = LQ LQ^T + 1e-4 I
        float s = (i4 == j4) ? 1e-4f : 0.0f;
        for (int k = 0; k < 16; k++) s += LQ[i4 * 16 + k] * LQ[j4 * 16 + k];
        Qs[i4][j4] = s;
    }
    if (tid < 64) {  // R = LR LR^T + 1e-4 I
        int i = tid >> 3, j = tid & 7;
        float s = (i == j) ? 1e-4f : 0.0f;
        for (int k = 0; k < 8; k++) s += LR[i * 8 + k] * LR[j * 8 + k];
        Rs[i][j] = s;
    }
    {   // P0 = L0 L0^T
        float s = 0.0f;
        for (int k = 0; k < 16; k++) s += l0elem(i4, k, ilog, ioff) * l0elem(j4, k, ilog, ioff);
        P[i4][j4] = s;
    }
    __syncthreads();
    {   // Hswz: H padded to 16x16, WMMA-A layout
        int q = tid >> 6, l = (tid >> 1) & 31, c = tid & 1;
        int m = l & 15, k = (q << 2) + ((l >> 4) << 1) + c;
        Hswz[(q * 32 + l) * 2 + c] = (m < 8) ? Hs[m][k] : 0.0f;
    }

    for (int t = 0; t < Tn; t++) {
        if (tid < 128) {  // HP = H P
            int m = tid >> 4, s = tid & 15;
            float a = 0.0f;
            for (int u = 0; u < 16; u++) a += Hs[m][u] * P[u][s];
            HP[m][s] = a;
        }
        __syncthreads();
        if (tid < 64) {  // Smat = HP H^T + R  == ycov_t
            int m = tid >> 3, n = tid & 7;
            float a = Rs[m][n];
            for (int s = 0; s < 16; s++) a += HP[m][s] * Hs[n][s];
            Sm[m][n] = a;
            covTab[t * 64 + tid] = a;
        }
        __syncthreads();
        if (t == Tn - 1) break;

        if (tid == 0) {  // Cholesky of 8x8 SPD Smat
            for (int i = 0; i < 8; i++)
                for (int j = 0; j <= i; j++) {
                    float s = Sm[i][j];
                    for (int k = 0; k < j; k++) s -= Lc[i][k] * Lc[j][k];
                    Lc[i][j] = (i == j) ? sqrtf(s) : s / Lc[j][j];
                }
        }
        __syncthreads();
        if (tid < 16) {  // solve Smat * Kt = HP, one RHS column per thread
            int s = tid;
            float y[8], x[8];
            for (int m = 0; m < 8; m++) {
                float a = HP[m][s];
                for (int k = 0; k < m; k++) a -= Lc[m][k] * y[k];
                y[m] = a / Lc[m][m];
            }
            for (int m = 7; m >= 0; m--) {
                float a = y[m];
                for (int k = m + 1; k < 8; k++) a -= Lc[k][m] * x[k];
                x[m] = a / Lc[m][m];
            }
            for (int m = 0; m < 8; m++) Kt[m][s] = x[m];
        }
        __syncthreads();
        {   // P_u = P - Kt^T HP
            float a = P[i4][j4];
            for (int m = 0; m < 8; m++) a -= Kt[m][i4] * HP[m][j4];
            Pu[i4][j4] = a;
        }
        if (tid < 128) {  // FK = F Kt^T   ([16,8])
            int i = tid >> 3, m = tid & 7;
            float a = 0.0f;
            for (int s = 0; s < 16; s++) a += Fs[i][s] * Kt[m][s];
            FK[i][m] = a;
        }
        __syncthreads();
        {   // FP = F P_u ; A_t = F - FK H
            float a = 0.0f;
            for (int s = 0; s < 16; s++) a += Fs[i4][s] * Pu[s][j4];
            FPm[i4][j4] = a;
            float b = Fs[i4][j4];
            for (int m = 0; m < 8; m++) b -= FK[i4][m] * Hs[m][j4];
            As[i4][j4] = b;
        }
        __syncthreads();
        {   // P_next = FP F^T + Q
            float a = Qs[i4][j4];
            for (int k = 0; k < 16; k++) a += FPm[i4][k] * Fs[j4][k];
            Pn[i4][j4] = a;
        }
        {   // emit A_t swizzled (4 chunks x 32 lanes x 2)
            int q = tid >> 6, l = (tid >> 1) & 31, c = tid & 1;
            int m = l & 15, k = (q << 2) + ((l >> 4) << 1) + c;
            Atab[((t * 4 + q) * 32 + l) * 2 + c] = As[m][k];
        }
        if (tid < 128) {  // emit B_t swizzled (2 chunks x 32 lanes x 2)
            int q = tid >> 6, l = (tid >> 1) & 31, c = tid & 1;
            int m = l & 15, k = (q << 2) + ((l >> 4) << 1) + c;
            Btab[((t * 2 + q) * 32 + l) * 2 + c] = FK[m][k];
        }
        __syncthreads();
        P[i4][j4] = Pn[i4][j4];
        __syncthreads();
    }
}

// ---------------------------------------------------------------------------
// Kernel 2: per-group mean recurrence, 16 groups per wave as WMMA N dimension.
// meanT_{t+1} = A_t meanT + B_t obsT ; ymT_t = Hpad meanT. Full-fp32 WMMA
// (V_WMMA_F32_16X16X4_F32), K chained in chunks of 4.
// ---------------------------------------------------------------------------
__global__ void __launch_bounds__(256)
kf_means(const float* __restrict__ obs, const float* __restrict__ mean0,
         const float* __restrict__ Atab, const float* __restrict__ Btab,
         const float* __restrict__ Hswz, float* __restrict__ outMeans) {
    __shared__ float ldsbuf[8][256];  // per-wave 16x16 transpose staging
    const int lane = threadIdx.x & 31;
    const int wib = threadIdx.x >> 5;
    const int wave = blockIdx.x * 8 + wib;
    const int g0 = wave * 16;
    const int h = lane >> 4, n = lane & 15;
    float* myLds = &ldsbuf[wib][0];

    v2f hA[4];
#pragma unroll
    for (int q = 0; q < 4; q++) hA[q] = *(const v2f*)(Hswz + (q * 32 + lane) * 2);

    // initial meanT in WMMA-B layout: meanT[s][g] = mean0[s] (all groups equal)
    v2f mB[4];
#pragma unroll
    for (int q = 0; q < 4; q++) {
        int k = q * 4 + h * 2;
        v2f v; v.x = mean0[k]; v.y = mean0[k + 1];
        mB[q] = v;
    }

    for (int t = 0; t < Tn; t++) {
        // ymT = Hpad * meanT  (rows 0..7 valid, live in lane-half 0)
        v8f ym = {};
#pragma unroll
        for (int q = 0; q < 4; q++)
            ym = __builtin_amdgcn_wmma_f32_16x16x4_f32(false, hA[q], false, mB[q],
                                                       (short)0, ym, false, false);
        if (h == 0) {
            float* dst = outMeans + ((size_t)(g0 + n) * Tn + t) * Mn;
#pragma unroll
            for (int j = 0; j < 8; j++) dst[j] = ym[j];
        }
        if (t == Tn - 1) break;

        // meanT_next = A_t * meanT + B_t * obsT
        v8f acc = {};
#pragma unroll
        for (int q = 0; q < 4; q++) {
            v2f a = *(const v2f*)(Atab + ((t * 4 + q) * 32 + lane) * 2);
            acc = __builtin_amdgcn_wmma_f32_16x16x4_f32(false, a, false, mB[q],
                                                        (short)0, acc, false, false);
        }
#pragma unroll
        for (int q = 0; q < 2; q++) {
            v2f b = *(const v2f*)(Btab + ((t * 2 + q) * 32 + lane) * 2);
            int k = q * 4 + h * 2;
            const float* op = obs + ((size_t)(g0 + n) * Tn + t) * Mn + k;
            v2f o; o.x = op[0]; o.y = op[1];
            acc = __builtin_amdgcn_wmma_f32_16x16x4_f32(false, b, false, o,
                                                        (short)0, acc, false, false);
        }

        // D-layout (lane holds col n, rows j+8h across 8 VGPRs) ->
        // B-layout (lane holds col n, rows 4q+2h,4q+2h+1 per chunk) via LDS.
        // DS ops are in-order within a wave; fence against compiler reordering.
#pragma unroll
        for (int j = 0; j < 8; j++) myLds[(j + 8 * h) * 16 + n] = acc[j];
        __builtin_amdgcn_wave_barrier();
        asm volatile("s_wait_dscnt 0" ::: "memory");
#pragma unroll
        for (int q = 0; q < 4; q++) {
            int k = q * 4 + h * 2;
            v2f v; v.x = myLds[k * 16 + n]; v.y = myLds[(k + 1) * 16 + n];
            mB[q] = v;
        }
        __builtin_amdgcn_wave_barrier();
    }
}

// ---------------------------------------------------------------------------
// Kernel 3: broadcast the 64KB cov table to [G,T,8,8] (134 MB) — HBM-bound.
// ---------------------------------------------------------------------------
__global__ void kf_covs(const float4* __restrict__ tab, float4* __restrict__ out, long n4) {
    long i = (long)blockIdx.x * blockDim.x + threadIdx.x;
    long stride = (long)gridDim.x * blockDim.x;
    for (; i < n4; i += stride) out[i] = tab[i & (Tn * 16 - 1)];
}

extern "C" void kernel_launch(void* const* d_in, const int* in_sizes, int n_in,
                              void* d_out, int out_size, void* d_ws, size_t ws_size,
                              hipStream_t stream) {
    const float* inp  = (const float*)d_in[0];  // [G,T,M]
    const float* m0   = (const float*)d_in[1];  // [S]
    const float* ilog = (const float*)d_in[2];  // [S]
    const float* ioff = (const float*)d_in[3];  // [S*(S-1)/2]
    const float* F    = (const float*)d_in[4];  // [S,S]
    const float* H    = (const float*)d_in[5];  // [M,S]
    const float* LQ   = (const float*)d_in[6];  // [S,S]
    const float* LR   = (const float*)d_in[7];  // [M,M]

    float* out = (float*)d_out;                         // means [G,T,M] then covs [G,T,M,M]
    float* outMeans = out;
    float* outCovs  = out + (size_t)Gn * Tn * Mn;

    float* ws = (float*)d_ws;                           // ~458 KB used
    float* covTab = ws;                                 // Tn*64        = 16384
    float* Atab   = covTab + Tn * 64;                   // (Tn-1)*256   = 65280
    float* Btab   = Atab + (Tn - 1) * 256;              // (Tn-1)*128   = 32640
    float* Hswz   = Btab + (Tn - 1) * 128;              // 256

    hipLaunchKernelGGL(kf_precompute, dim3(1), dim3(256), 0, stream,
                       ilog, ioff, F, H, LQ, LR, covTab, Atab, Btab, Hswz);

    hipLaunchKernelGGL(kf_means, dim3(Gn / 16 / 8), dim3(256), 0, stream,
                       inp, m0, Atab, Btab, Hswz, outMeans);

    long n4 = (long)Gn * Tn * 16;  // covs as float4s
    hipLaunchKernelGGL(kf_covs, dim3(4096), dim3(256), 0, stream,
                       (const float4*)covTab, (float4*)outCovs, n4);
}